// Text3DSpace_85306640433865
// MI455X (gfx1250) — compile-verified
//
#include <hip/hip_runtime.h>
#include <hip/hip_bf16.h>
#include <cstdint>

// ---------------- problem constants (from setup_inputs) ----------------
#define L_VAL   6
#define GRIDW   13            // 2L+1
#define NHIP    30            // n_hiper
#define HTOT    31            // n_hiper + 1
#define DCH     768           // channel dim
#define NSAMP   2048
#define NCH     32            // n-chunks (segmented scan)
#define CHUNK   (NSAMP / NCH) // 64 samples per chunk
#define DTILE   64            // d-channels per block (1 per lane, 2 waves)
#define NDT     (DCH / DTILE) // 12 d-tiles
#define CELLSTRIDE (HTOT * DCH)      // 23808 floats per grid cell
#define TILEFLTS (8 * HTOT * DTILE)  // floats per corner-tile buffer (one sample)
#define LDS_BYTES (2 * TILEFLTS * 4) // double buffer: 126976 B (<320KB/WGP)

// ---------------- CDNA5 TDM path selection ----------------
#if defined(__HIP_DEVICE_COMPILE__) && __has_builtin(__builtin_amdgcn_tensor_load_to_lds)
  #define USE_TDM 1
#else
  #define USE_TDM 0
#endif

#if USE_TDM
typedef unsigned int tdm_u32x4 __attribute__((ext_vector_type(4)));
typedef int          tdm_i32x4 __attribute__((ext_vector_type(4)));
typedef int          tdm_i32x8 __attribute__((ext_vector_type(8)));
#if __has_include(<hip/amd_detail/amd_gfx1250_TDM.h>)
  #define TDM_ARGS6 1   // therock-10.0 toolchain: 6-arg builtin
#else
  #define TDM_ARGS6 0   // ROCm 7.2 clang-22: 5-arg builtin
#endif

// Issue one TDM 2D-tile load: HTOT rows x DTILE f32, row stride DCH floats,
// global -> LDS. Descriptor fields per CDNA5 ISA ch.8 (D# groups 0/1).
__device__ __forceinline__ void tdm_load_tile(uint32_t lds_addr, unsigned long long gaddr) {
    tdm_u32x4 g0;
    g0[0] = 1u;                                            // count=1, user mode
    g0[1] = lds_addr;                                      // LDS byte address
    g0[2] = (uint32_t)(gaddr & 0xFFFFFFFFull);             // global_addr[31:0]
    g0[3] = (uint32_t)((gaddr >> 32) & 0x1FFFFFFull)       // global_addr[56:32]
          | (2u << 30);                                    // type=2 ("image")
    tdm_i32x8 g1;
    g1[0] = (2 << 16);            // workgroup_mask=0, data_size=2 (4 bytes)
    g1[1] = (DCH & 0xFFFF) << 16; // tensor_dim0[15:0] in bits[63:48]
    g1[2] = (HTOT << 16);         // tensor_dim0[31:16]=0 | tensor_dim1[15:0]=31
    g1[3] = (DTILE << 16);        // tensor_dim1[31:16]=0 | tile_dim0=64
    g1[4] = HTOT;                 // tile_dim1=31 | tile_dim2=0
    g1[5] = DCH;                  // tensor_dim0_stride[31:0] = 768 elements
    g1[6] = 0;                    // stride0 hi | tensor_dim1_stride lo
    g1[7] = 0;
    tdm_i32x4 z4 = {0, 0, 0, 0};  // groups 2/3 unused (2D tensor)
#if TDM_ARGS6
    tdm_i32x8 z8 = {0, 0, 0, 0, 0, 0, 0, 0};
    __builtin_amdgcn_tensor_load_to_lds(g0, g1, z4, z4, z8, 0);
#else
    __builtin_amdgcn_tensor_load_to_lds(g0, g1, z4, z4, 0);
#endif
}

// Issue the 8 corner-tile DMAs of one sample into LDS buffer `buf`.
__device__ __forceinline__ void issue_corners(const float* emb, const int* cell,
                                              int d0, const float* buf) {
#pragma unroll
    for (int c = 0; c < 8; ++c) {
        unsigned long long ga = (unsigned long long)(uintptr_t)
            (emb + (size_t)cell[c] * CELLSTRIDE + d0);
        uint32_t la = (uint32_t)(uintptr_t)(buf + c * (HTOT * DTILE));
        tdm_load_tile(la, ga);
    }
}
#endif // USE_TDM

// Trilinear cell indices + corner weights for sample n (uniform per block).
__device__ __forceinline__ void cells_weights(const float* __restrict__ samples,
                                              int n, float* wc, int* cell) {
    const float sx = samples[3 * n + 0] + (float)L_VAL;
    const float sy = samples[3 * n + 1] + (float)L_VAL;
    const float sz = samples[3 * n + 2] + (float)L_VAL;
    const float fx = floorf(sx), fy = floorf(sy), fz = floorf(sz);
    const float dx = sx - fx, dy = sy - fy, dz = sz - fz;
    const int ix0 = (int)fx, iy0 = (int)fy, iz0 = (int)fz;
    const int ix1 = min(ix0 + 1, GRIDW - 1);
    const int iy1 = min(iy0 + 1, GRIDW - 1);
    const int iz1 = min(iz0 + 1, GRIDW - 1);
    const float wx[2] = {1.0f - dx, dx};
    const float wy[2] = {1.0f - dy, dy};
    const float wz[2] = {1.0f - dz, dz};
    const int   xi[2] = {ix0, ix1}, yi[2] = {iy0, iy1}, zi[2] = {iz0, iz1};
#pragma unroll
    for (int c = 0; c < 8; ++c) {
        const int xs = c & 1, ys = (c >> 1) & 1, zs = (c >> 2) & 1;
        wc[c]   = wx[xs] * wy[ys] * wz[zs];
        cell[c] = (xi[xs] * GRIDW + yi[ys]) * GRIDW + zi[zs];
    }
}

// ---------------- kernel 1: per-sample segment lengths ----------------
__global__ __launch_bounds__(256) void t3d_dists_kernel(
    const float* __restrict__ samples, const float* __restrict__ last_point,
    float* __restrict__ dists)
{
    int n = blockIdx.x * 256 + threadIdx.x;
    if (n >= NSAMP) return;
    float ax = samples[3 * n + 0], ay = samples[3 * n + 1], az = samples[3 * n + 2];
    float bx, by, bz;
    if (n == NSAMP - 1) {
        bx = last_point[0]; by = last_point[1]; bz = last_point[2];
    } else {
        bx = samples[3 * n + 3]; by = samples[3 * n + 4]; bz = samples[3 * n + 5];
    }
    float ddx = bx - ax, ddy = by - ay, ddz = bz - az;
    dists[n] = sqrtf(ddx * ddx + ddy * ddy + ddz * ddz);
}

// ---------------- kernel 2: fused trilinear + local volume render ----------------
// grid = (NDT d-tiles, NCH chunks); block = DTILE lanes, 1 channel each.
// Double-buffered TDM pipeline: while computing sample k from buffer (k&1), the
// 8 corner DMAs of sample k+1 are in flight into buffer ((k+1)&1); TDM
// completion is in-order per wave, so s_wait_tensorcnt<=8 guarantees buffer k
// is resident while the next batch stays outstanding.
__global__ __launch_bounds__(DTILE) void t3d_pass1_kernel(
    const float* __restrict__ samples, const float* __restrict__ emb,
    const float* __restrict__ dists, float* __restrict__ wsT, float* __restrict__ wsS)
{
    const int tid   = threadIdx.x;
    const int d0    = blockIdx.x * DTILE;
    const int d     = d0 + tid;
    const int chunk = blockIdx.y;
    const int n0    = chunk * CHUNK;

#if USE_TDM
    extern __shared__ float smem[];  // 2 * TILEFLTS floats (dynamic LDS)
    // NOTE: select buffer halves with pointer arithmetic only — putting these
    // pointers into an aggregate initializer creates an addrspacecast constant
    // that the device linker rejects.
#endif

    float S[NHIP];
#pragma unroll
    for (int h = 0; h < NHIP; ++h) S[h] = 0.0f;
    float Tloc = 1.0f;

    float wc[8];
    int   cell[8];
    cells_weights(samples, n0, wc, cell);

#if USE_TDM
    if (tid == 0) issue_corners(emb, cell, d0, smem);  // prologue DMA for k=0
#endif

    for (int k = 0; k < CHUNK; ++k) {
        const int n = n0 + k;

        float wc_n[8];
        int   cell_n[8];
        if (k + 1 < CHUNK) cells_weights(samples, n + 1, wc_n, cell_n);

#if USE_TDM
        if (tid == 0) {
            if (k + 1 < CHUNK) {
                issue_corners(emb, cell_n, d0, smem + ((k + 1) & 1) * TILEFLTS);
                __builtin_amdgcn_s_wait_tensorcnt(8);  // batch k done; k+1 in flight
            } else {
                __builtin_amdgcn_s_wait_tensorcnt(0);
            }
        }
        __syncthreads();                                  // publish buffer k
        const float* buf = smem + (k & 1) * TILEFLTS;
#endif

        float interp[HTOT];
#pragma unroll
        for (int h = 0; h < HTOT; ++h) interp[h] = 0.0f;

#if USE_TDM
#pragma unroll
        for (int c = 0; c < 8; ++c) {
            const float w = wc[c];
            const float* p = buf + c * (HTOT * DTILE) + tid;
#pragma unroll
            for (int h = 0; h < HTOT; ++h)
                interp[h] = fmaf(w, p[h * DTILE], interp[h]);
        }
        __syncthreads();  // all reads of buffer k done before it is re-DMAed (k+2)
#else
        // Fallback: direct coalesced gathers (d across lanes, rows stride 768).
#pragma unroll
        for (int c = 0; c < 8; ++c) {
            const float w = wc[c];
            const float* p = emb + (size_t)cell[c] * CELLSTRIDE + d;
#pragma unroll
            for (int h = 0; h < HTOT; ++h)
                interp[h] = fmaf(w, p[h * DCH], interp[h]);
        }
#endif

        const float density = fmaxf(interp[NHIP], 0.0f);
        const float dd      = density * dists[n];
        const float e       = __expf(-dd);
        const float a       = Tloc * (1.0f - e); // exp(-local_excl)*(1-exp(-dd))
#pragma unroll
        for (int h = 0; h < NHIP; ++h) S[h] = fmaf(a, interp[h], S[h]);
        Tloc *= e;

#pragma unroll
        for (int c = 0; c < 8; ++c) { wc[c] = wc_n[c]; cell[c] = cell_n[c]; }
    }

    wsT[(size_t)chunk * DCH + d] = Tloc;
#pragma unroll
    for (int h = 0; h < NHIP; ++h)
        wsS[((size_t)chunk * NHIP + h) * DCH + d] = S[h];
}

// ---------------- kernel 3: combine chunks (carry transmittance prefix) ----------------
__global__ __launch_bounds__(128) void t3d_pass2_kernel(
    const float* __restrict__ wsT, const float* __restrict__ wsS,
    float* __restrict__ out)
{
    const int d = blockIdx.x * 128 + threadIdx.x; // 6 blocks x 128 = 768
    float acc[NHIP];
#pragma unroll
    for (int h = 0; h < NHIP; ++h) acc[h] = 0.0f;
    float T = 1.0f; // exp(-excl) at chunk start
    for (int j = 0; j < NCH; ++j) {
        const float t = wsT[(size_t)j * DCH + d];
#pragma unroll
        for (int h = 0; h < NHIP; ++h)
            acc[h] = fmaf(T, wsS[((size_t)j * NHIP + h) * DCH + d], acc[h]);
        T *= t;
    }
#pragma unroll
    for (int h = 0; h < NHIP; ++h)
        out[(size_t)h * DCH + d] = acc[h];
}

// ---------------- launch ----------------
extern "C" void kernel_launch(void* const* d_in, const int* in_sizes, int n_in,
                              void* d_out, int out_size, void* d_ws, size_t ws_size,
                              hipStream_t stream) {
    const float* samples    = (const float*)d_in[0]; // (2048,3)
    const float* last_point = (const float*)d_in[1]; // (3,)
    const float* emb        = (const float*)d_in[2]; // (13,13,13,31,768)
    // d_in[3] = layers_num (6), d_in[4] = n_hiper (30): baked into constants.

    float* ws    = (float*)d_ws;
    float* dists = ws;                         // 2048 floats
    float* wsT   = dists + NSAMP;              // NCH*768 floats
    float* wsS   = wsT + (size_t)NCH * DCH;    // NCH*30*768 floats (~2.9 MB total)
    float* out   = (float*)d_out;              // (30,768)

    t3d_dists_kernel<<<dim3(NSAMP / 256), dim3(256), 0, stream>>>(samples, last_point, dists);
    t3d_pass1_kernel<<<dim3(NDT, NCH), dim3(DTILE), LDS_BYTES, stream>>>(samples, emb, dists, wsT, wsS);
    t3d_pass2_kernel<<<dim3(DCH / 128), dim3(128), 0, stream>>>(wsT, wsS, out);
}